// GraphConvLayer_pyT_22351009808763
// MI455X (gfx1250) — compile-verified
//
#include <hip/hip_runtime.h>
#include <hip/hip_bf16.h>
#include <stdint.h>

// Problem constants (match reference)
#define NN     8192        // nodes
#define EE     262144      // edges
#define DD     128         // feature dim (in == out)
#define WPR    256         // bitmask words per row (8192 / 32)

typedef __attribute__((ext_vector_type(2))) float v2f;
typedef __attribute__((ext_vector_type(8))) float v8f;

// ---------------------------------------------------------------------------
// 1) zero the adjacency bitmask (8192 rows x 256 u32 = 8 MB)
// ---------------------------------------------------------------------------
__global__ void k_zero_mask(uint32_t* __restrict__ mask) {
    int i = blockIdx.x * blockDim.x + threadIdx.x;   // 0 .. NN*WPR-1
    mask[i] = 0u;
}

// ---------------------------------------------------------------------------
// 2) set diagonal (self-loop) bits; runs after zeroing, before scatter.
//    Word i*WPR + (i>>5) is unique per thread -> plain store of the bit.
// ---------------------------------------------------------------------------
__global__ void k_diag(uint32_t* __restrict__ mask) {
    int i = blockIdx.x * blockDim.x + threadIdx.x;   // 0 .. NN-1
    mask[i * WPR + (i >> 5)] = (1u << (i & 31));
}

// ---------------------------------------------------------------------------
// 3) scatter edges: atomicOr dedupes duplicate edges exactly like the
//    reference's .at[...].set(1.0)
// ---------------------------------------------------------------------------
__global__ void k_scatter(const int* __restrict__ idx, uint32_t* __restrict__ mask) {
    int e = blockIdx.x * blockDim.x + threadIdx.x;   // 0 .. EE-1
    int r = idx[e];
    int c = idx[EE + e];
    atomicOr(&mask[r * WPR + (c >> 5)], 1u << (c & 31));
}

// ---------------------------------------------------------------------------
// 4) degree = popcount(row bits); norm = deg^-1/2  (deg >= 1 due to self loop)
// ---------------------------------------------------------------------------
__global__ void k_norm(const uint32_t* __restrict__ mask, float* __restrict__ norm) {
    int i = blockIdx.x * blockDim.x + threadIdx.x;   // 0 .. NN-1
    const uint4* row = (const uint4*)(mask + (size_t)i * WPR);
    int cnt = 0;
#pragma unroll 4
    for (int w = 0; w < WPR / 4; ++w) {
        uint4 v = row[w];
        cnt += __popc(v.x) + __popc(v.y) + __popc(v.z) + __popc(v.w);
    }
    norm[i] = rsqrtf((float)cnt);
}

// ---------------------------------------------------------------------------
// 5) hs = diag(norm) * (x @ W^T + b)   via V_WMMA_F32_16X16X4_F32
//    One wave computes one 16x16 tile; K=128 -> 32 WMMA ops per wave.
//    Tiles: 512 (M) x 8 (N) = 4096 waves = 1024 blocks of 128 threads.
//
//    A (16x4 fp32, 2 VGPRs): lanes 0-15 carry M=lane, K={0,1};
//                            lanes 16-31 carry M=lane-16, K={2,3}.
//    B (4x16 fp32, 2 VGPRs): mirrored (N from lane, K split by lane half).
//    C/D (16x16 fp32, 8 VGPRs): VGPR r -> M = r + 8*(lane>=16), N = lane&15.
//
//    Source-side norm is folded into the epilogue: norm[m] is half-wave
//    uniform (8 broadcast loads/wave), removing one gather + one multiply
//    per neighbor per feature from the hot aggregation loop.
// ---------------------------------------------------------------------------
__global__ void k_gemm_wmma(const float* __restrict__ x,
                            const float* __restrict__ W,
                            const float* __restrict__ bias,
                            const float* __restrict__ norm,
                            float* __restrict__ hs) {
    int wave = blockIdx.x * (blockDim.x >> 5) + (threadIdx.x >> 5);
    int lane = threadIdx.x & 31;
    int half = lane >> 4;      // 0: K offset 0, 1: K offset 2
    int l16  = lane & 15;

    int wm = wave >> 3;        // 0..511  M-tile
    int wn = wave & 7;         // 0..7    N-tile

    int arow = wm * 16 + l16;  // x row feeding this lane's A slots
    int bcol = wn * 16 + l16;  // W row (output feature) feeding this lane's B slots

    // bias folded into accumulator init: every (M, N=bcol) element starts at b[N]
    float bv = bias[bcol];
    v8f c;
#pragma unroll
    for (int r = 0; r < 8; ++r) c[r] = bv;

    const float* xr = x + (size_t)arow * DD + 2 * half;
    const float* wr = W + (size_t)bcol * DD + 2 * half;

#pragma unroll
    for (int k = 0; k < DD; k += 4) {
        v2f a  = *(const v2f*)(xr + k);   // K = k+2*half, k+2*half+1
        v2f bm = *(const v2f*)(wr + k);
        c = __builtin_amdgcn_wmma_f32_16x16x4_f32(
                /*neg_a=*/false, a, /*neg_b=*/false, bm,
                /*c_mod=*/(short)0, c, /*reuse_a=*/false, /*reuse_b=*/false);
    }

#pragma unroll
    for (int r = 0; r < 8; ++r) {
        int m = wm * 16 + half * 8 + r;
        hs[(size_t)m * DD + wn * 16 + l16] = norm[m] * c[r];
    }
}

// ---------------------------------------------------------------------------
// 6) out[i,:] = n_i * sum_{j : bit(i,j)} hs[j,:]
//    One block per row, 128 threads = one feature each. Mask words staged in
//    LDS; bit-walk is wave-uniform so branches go scalar. hs row gathers are
//    fully coalesced 512B transactions and hit L2 (hs is 4 MB).
// ---------------------------------------------------------------------------
__global__ void k_aggregate(const uint32_t* __restrict__ mask,
                            const float* __restrict__ norm,
                            const float* __restrict__ hs,
                            float* __restrict__ out) {
    __shared__ uint32_t words[WPR];
    int i = blockIdx.x;        // row
    int t = threadIdx.x;       // feature 0..127

    words[t]       = mask[(size_t)i * WPR + t];
    words[t + 128] = mask[(size_t)i * WPR + 128 + t];
    __syncthreads();

    float acc = 0.0f;
    for (int w = 0; w < WPR; ++w) {
        uint32_t m = words[w];
        while (m) {
            int b = __ffs(m) - 1;
            m &= m - 1;
            int j = (w << 5) + b;
            acc += hs[(size_t)j * DD + t];
        }
    }
    out[(size_t)i * DD + t] = norm[i] * acc;
}

// ---------------------------------------------------------------------------
// launcher
// ---------------------------------------------------------------------------
extern "C" void kernel_launch(void* const* d_in, const int* in_sizes, int n_in,
                              void* d_out, int out_size, void* d_ws, size_t ws_size,
                              hipStream_t stream) {
    const float* x    = (const float*)d_in[0];
    const int*   eidx = (const int*)d_in[1];       // [2, E] flat: row0 then row1
    const float* W    = (const float*)d_in[2];     // [D_OUT, D_IN]
    const float* bias = (const float*)d_in[3];     // [D_OUT]
    float*       out  = (float*)d_out;

    // workspace carve-up: bitmask (8 MB) | norm (32 KB) | hs (4 MB)
    uint32_t* mask = (uint32_t*)d_ws;
    float*    norm = (float*)(mask + (size_t)NN * WPR);
    float*    hs   = norm + NN;

    k_zero_mask<<<(NN * WPR) / 256, 256, 0, stream>>>(mask);
    k_diag<<<NN / 256, 256, 0, stream>>>(mask);
    k_scatter<<<EE / 256, 256, 0, stream>>>(eidx, mask);
    k_norm<<<NN / 256, 256, 0, stream>>>(mask, norm);
    k_gemm_wmma<<<(NN / 16) * (DD / 16) / 4, 128, 0, stream>>>(x, W, bias, norm, hs);
    k_aggregate<<<NN, 128, 0, stream>>>(mask, norm, hs, out);
}